// IGNet_42099269436028
// MI455X (gfx1250) — compile-verified
//
#include <hip/hip_runtime.h>
#include <hip/hip_bf16.h>
#include <stdint.h>

// IGNet / SAGE-mean x3 on MI455X (gfx1250).
// L2-resident scatter/gather workload: feature table (25.6 MB) + edge lists
// (6.4 MB) all fit in the 192 MB L2. wave32: one wave per edge, one float4
// per lane (128 floats / 32 lanes). Edge-index chunks are staged into LDS by
// the Tensor Data Mover (tensor_load_to_lds, TENSORcnt), gathers are
// prefetched (global_prefetch_b8), and the segment-sum uses f32 L2 atomics.

#define DIM 128
#define WAVES_PER_BLOCK 8
#define EDGE_CHUNK 256   // edges staged per block via TDM (1 KB per index array)

typedef unsigned int u32x4 __attribute__((ext_vector_type(4)));
typedef int          i32x8 __attribute__((ext_vector_type(8)));
typedef int          i32x4 __attribute__((ext_vector_type(4)));

// Build a 1-D TDM descriptor (D#) and issue TENSOR_LOAD_TO_LDS.
// Layout per CDNA5 ISA 8.3/8.4:
//  group0: [1:0]=count=1, [63:32]=lds_addr, [120:64]=global_addr, [127:126]=type=2
//  group1: [17:16]=data_size(2 => 4B), [79:48]=tensor_dim0 (OOB reads -> 0),
//          [111:80]=tensor_dim1=1, [127:112]=tile_dim0, strides benign.
// groups 2/3 + extra group zero (dims <= 2D). Flat address of LDS: low 32 bits
// are the LDS byte offset (ISA 10.2 aperture mapping), so &shared[0] truncated
// works. This toolchain's builtin is the 6-arg form:
//   (u32x4 g0, i32x8 g1, i32x4 g2, i32x4 g3, i32x8 extra, i32 cpol)
__device__ __forceinline__ void tdm_load_i32_1d(unsigned lds_byte_addr,
                                                const int* gsrc,
                                                unsigned tensor_len,
                                                unsigned tile_len) {
  unsigned long long ga = (unsigned long long)(uintptr_t)gsrc;
  u32x4 g0 = { 1u,                                    // count=1 (valid user D#)
               lds_byte_addr,                         // LDS dest (bytes)
               (unsigned)(ga & 0xFFFFFFFFull),        // global_addr[31:0]
               (unsigned)((ga >> 32) & 0x01FFFFFFull) // global_addr[56:32]
                   | (2u << 30) };                    // type = 2 ("image")
  i32x8 g1 = { (int)(2u << 16),                            // data_size = 4 bytes
               (int)((tensor_len & 0xFFFFu) << 16),        // tensor_dim0[15:0]
               (int)(((tensor_len >> 16) & 0xFFFFu)        // tensor_dim0[31:16]
                     | (1u << 16)),                        // tensor_dim1 = 1
               (int)((tile_len & 0xFFFFu) << 16),          // tile_dim0
               0,                                          // tile_dim1/2 unused
               (int)tensor_len,                            // dim0 stride (benign)
               0, 0 };
  i32x4 gz4 = { 0, 0, 0, 0 };
  i32x8 gz8 = { 0, 0, 0, 0, 0, 0, 0, 0 };
  __builtin_amdgcn_tensor_load_to_lds(g0, g1, gz4, gz4, gz8, 0);
}

__global__ void zero_kernel(float* __restrict__ p, int n) {
  int i = blockIdx.x * blockDim.x + threadIdx.x;
  if (i < n) p[i] = 0.0f;
}

__global__ void deg_accum_kernel(const int* __restrict__ edge_dst,
                                 float* __restrict__ deg, int E) {
  int e = blockIdx.x * blockDim.x + threadIdx.x;
  if (e < E) atomicAdd(&deg[edge_dst[e]], 1.0f);
}

__global__ void deg_finalize_kernel(const float* __restrict__ deg,
                                    float* __restrict__ deg_inv, int N) {
  int i = blockIdx.x * blockDim.x + threadIdx.x;
  if (i < N) deg_inv[i] = 1.0f / fmaxf(deg[i], 1.0f);
}

// res = h; tmp = h  (float4 vectorized)
__global__ void init_copy_kernel(const float* __restrict__ h,
                                 float* __restrict__ tmp,
                                 float* __restrict__ res, int n4) {
  int i = blockIdx.x * blockDim.x + threadIdx.x;
  if (i < n4) {
    const float4 v = reinterpret_cast<const float4*>(h)[i];
    reinterpret_cast<float4*>(tmp)[i] = v;
    reinterpret_cast<float4*>(res)[i] = v;
  }
}

// Segment-sum scatter: one wave per edge, lane -> float4 column chunk.
// Edge indices for this block are DMA'd into LDS by the TDM (wave 0 issues,
// TENSORcnt waited, then workgroup barrier publishes LDS to all 8 waves).
__global__ void __launch_bounds__(WAVES_PER_BLOCK * 32)
scatter_kernel(const float* __restrict__ tmp,
               const int* __restrict__ edge_src,
               const int* __restrict__ edge_dst,
               float* __restrict__ agg, int E) {
  __shared__ int s_src[EDGE_CHUNK];
  __shared__ int s_dst[EDGE_CHUNK];

  const int base = blockIdx.x * EDGE_CHUNK;
  if (threadIdx.x == 0) {  // TDM ignores EXEC; one issue per array from wave 0
    const unsigned remaining = (unsigned)(E - base);  // tail OOB -> zero-fill
    tdm_load_i32_1d((unsigned)(uintptr_t)&s_src[0], edge_src + base,
                    remaining, EDGE_CHUNK);
    tdm_load_i32_1d((unsigned)(uintptr_t)&s_dst[0], edge_dst + base,
                    remaining, EDGE_CHUNK);
    __builtin_amdgcn_s_wait_tensorcnt(0);
  }
  __syncthreads();

  const int wave = threadIdx.x >> 5;
  const int lane = threadIdx.x & 31;
  const int n = min(EDGE_CHUNK, E - base);

  for (int e = wave; e < n; e += WAVES_PER_BLOCK) {
    const int nx = e + WAVES_PER_BLOCK;
    if (nx < n)  // hide L2 latency of the next gather (global_prefetch_b8)
      __builtin_prefetch(tmp + (size_t)s_src[nx] * DIM + lane * 4, 0, 0);

    const float4 v = *reinterpret_cast<const float4*>(
        tmp + (size_t)s_src[e] * DIM + lane * 4);         // 512 B/wave, coalesced
    float* a = agg + (size_t)s_dst[e] * DIM + lane * 4;   // L2-resident atomics
    atomicAdd(a + 0, v.x);
    atomicAdd(a + 1, v.y);
    atomicAdd(a + 2, v.z);
    atomicAdd(a + 3, v.w);
  }
}

// tmp += agg * deg_inv[node]; res += tmp * coef   (float4 vectorized)
__global__ void combine_kernel(const float* __restrict__ agg,
                               const float* __restrict__ deg_inv,
                               float* __restrict__ tmp,
                               float* __restrict__ res,
                               float coef, int n4) {
  int i = blockIdx.x * blockDim.x + threadIdx.x;
  if (i >= n4) return;
  const int node = i >> 5;  // 32 float4 chunks per 128-wide row
  const float di = deg_inv[node];
  const float4 a = reinterpret_cast<const float4*>(agg)[i];
  float4 t = reinterpret_cast<float4*>(tmp)[i];
  t.x += a.x * di; t.y += a.y * di; t.z += a.z * di; t.w += a.w * di;
  reinterpret_cast<float4*>(tmp)[i] = t;
  float4 r = reinterpret_cast<float4*>(res)[i];
  r.x += t.x * coef; r.y += t.y * coef; r.z += t.z * coef; r.w += t.w * coef;
  reinterpret_cast<float4*>(res)[i] = r;
}

extern "C" void kernel_launch(void* const* d_in, const int* in_sizes, int n_in,
                              void* d_out, int out_size, void* d_ws, size_t ws_size,
                              hipStream_t stream) {
  (void)n_in; (void)out_size; (void)ws_size;
  const float* h    = (const float*)d_in[0];
  const int*   esrc = (const int*)d_in[1];
  const int*   edst = (const int*)d_in[2];
  const int N = in_sizes[0] / DIM;
  const int E = in_sizes[1];

  float* res = (float*)d_out;

  // Workspace: deg[N] | deg_inv[N] | agg[N*DIM] | tmp[N*DIM]  (~51.5 MB)
  float* ws      = (float*)d_ws;
  float* deg     = ws;
  float* deg_inv = ws + N;
  float* agg     = ws + 2 * (size_t)N;
  float* tmp     = agg + (size_t)N * DIM;

  const int nd  = N * DIM;
  const int n4  = nd / 4;
  const int TB  = 256;

  zero_kernel<<<(N + TB - 1) / TB, TB, 0, stream>>>(deg, N);
  deg_accum_kernel<<<(E + TB - 1) / TB, TB, 0, stream>>>(edst, deg, E);
  deg_finalize_kernel<<<(N + TB - 1) / TB, TB, 0, stream>>>(deg, deg_inv, N);
  init_copy_kernel<<<(n4 + TB - 1) / TB, TB, 0, stream>>>(h, tmp, res, n4);

  const float coefs[3] = {1.0f, 0.5f, 1.0f / 3.0f};
  const int edge_blocks = (E + EDGE_CHUNK - 1) / EDGE_CHUNK;
  for (int l = 0; l < 3; ++l) {
    zero_kernel<<<(nd + TB - 1) / TB, TB, 0, stream>>>(agg, nd);
    scatter_kernel<<<edge_blocks, WAVES_PER_BLOCK * 32, 0, stream>>>(
        tmp, esrc, edst, agg, E);
    combine_kernel<<<(n4 + TB - 1) / TB, TB, 0, stream>>>(
        agg, deg_inv, tmp, res, coefs[l], n4);
  }
}